// MixtureOfExperts_7464653160759
// MI455X (gfx1250) — compile-verified
//
#include <hip/hip_runtime.h>
#include <math.h>

#define D_MODEL 1024
#define FF      4096
#define NEXP    8
#define NTOK    4096
#define TILE_M  16
#define CHUNK   128          // d_ff chunk = 8 waves x 16 cols
#define XS_STRIDE 1028       // 1028 % 64 == 4 -> conflict-free row reads
#define HS_STRIDE 132        // 132 % 64 == 4

typedef __attribute__((ext_vector_type(2))) float v2f;
typedef __attribute__((ext_vector_type(8))) float v8f;

// ---------------- gating: logits -> top2 -> softmax -> per-expert lists ----
__global__ __launch_bounds__(256) void moe_gate_kernel(
    const float* __restrict__ x, const float* __restrict__ gw,
    int* __restrict__ counts, int* __restrict__ tok_id, float* __restrict__ tok_w)
{
  const int token = blockIdx.x * 8 + (threadIdx.x >> 5);
  const int lane  = threadIdx.x & 31;
  if (token >= NTOK) return;

  const float* xr = x + (size_t)token * D_MODEL;
  float acc[NEXP];
#pragma unroll
  for (int e = 0; e < NEXP; ++e) acc[e] = 0.f;
  for (int d = lane; d < D_MODEL; d += 32) {
    const float xv = xr[d];
    const float* g = gw + (size_t)d * NEXP;
#pragma unroll
    for (int e = 0; e < NEXP; ++e) acc[e] += xv * g[e];
  }
#pragma unroll
  for (int e = 0; e < NEXP; ++e) {
    float v = acc[e];
#pragma unroll
    for (int off = 16; off > 0; off >>= 1) v += __shfl_xor(v, off, 32);
    acc[e] = v;
  }
  if (lane == 0) {
    int i0 = 0; float v0 = acc[0];
#pragma unroll
    for (int e = 1; e < NEXP; ++e) if (acc[e] > v0) { v0 = acc[e]; i0 = e; }
    int i1 = -1; float v1 = -3.4e38f;
#pragma unroll
    for (int e = 0; e < NEXP; ++e) if (e != i0 && acc[e] > v1) { v1 = acc[e]; i1 = e; }
    const float ex = expf(v1 - v0);          // v0 >= v1
    const float s  = 1.f + ex;
    const float w0 = 1.f / s;
    const float w1 = ex / s;
    int s0 = atomicAdd(&counts[i0], 1);
    tok_id[i0 * NTOK + s0] = token; tok_w[i0 * NTOK + s0] = w0;
    int s1 = atomicAdd(&counts[i1], 1);
    tok_id[i1 * NTOK + s1] = token; tok_w[i1 * NTOK + s1] = w1;
  }
}

// ---------------- fused expert MLP: out += wt * (gelu(x@w1+b1) @ w2 + b2) ----
__global__ __launch_bounds__(256) void moe_expert_kernel(
    const float* __restrict__ x,
    const float* __restrict__ w1, const float* __restrict__ b1,
    const float* __restrict__ w2, const float* __restrict__ b2,
    const int* __restrict__ counts, const int* __restrict__ tok_id,
    const float* __restrict__ tok_w, float* __restrict__ out)
{
  __shared__ float xs[TILE_M * XS_STRIDE];   // 16 x 1024 token tile
  __shared__ float hs[TILE_M * HS_STRIDE];   // 16 x 128 gelu chunk
  __shared__ int   s_tok[TILE_M];
  __shared__ float s_wt[TILE_M];

  const int e   = blockIdx.y;
  const int cnt = counts[e];                 // uniform across block
  const int t0  = blockIdx.x * TILE_M;
  if (t0 >= cnt) return;                     // uniform exit

  const int tid  = threadIdx.x;
  const int wave = tid >> 5;                 // 0..7
  const int lane = tid & 31;
  const int half = lane >> 4;                // 0 | 1
  const int l16  = lane & 15;

  if (tid < TILE_M) {
    const int idx = t0 + tid;
    if (idx < cnt) { s_tok[tid] = tok_id[e * NTOK + idx]; s_wt[tid] = tok_w[e * NTOK + idx]; }
    else           { s_tok[tid] = tok_id[e * NTOK + t0];  s_wt[tid] = 0.0f; }
  }
  __syncthreads();

  // stage x tile into LDS (coalesced float4 row copies)
  for (int i = tid; i < TILE_M * (D_MODEL / 4); i += 256) {
    const int row = i / (D_MODEL / 4);
    const int c4  = i % (D_MODEL / 4);
    const float4 v = reinterpret_cast<const float4*>(x + (size_t)s_tok[row] * D_MODEL)[c4];
    float* dst = &xs[row * XS_STRIDE + c4 * 4];
    dst[0] = v.x; dst[1] = v.y; dst[2] = v.z; dst[3] = v.w;
  }
  __syncthreads();

  const float* w1e = w1 + (size_t)e * D_MODEL * FF;
  const float* w2e = w2 + (size_t)e * FF * D_MODEL;
  const float* b1e = b1 + (size_t)e * FF;
  const float* b2e = b2 + (size_t)e * D_MODEL;

  const v8f zero8 = {0.f,0.f,0.f,0.f,0.f,0.f,0.f,0.f};
  v8f oacc[8];
#pragma unroll
  for (int j = 0; j < 8; ++j) oacc[j] = zero8;

  for (int ch = 0; ch < FF / CHUNK; ++ch) {
    // ---- stage 1: h(16 x 16) for this wave's columns, K over full D_MODEL ----
    const int c0 = ch * CHUNK + wave * 16;
    v8f hacc = zero8;
    const float* w1c = w1e + c0 + l16;
    for (int kk = 0; kk < D_MODEL; kk += 4) {
      v2f a;
      const float* ap = &xs[l16 * XS_STRIDE + kk + 2 * half];
      a[0] = ap[0]; a[1] = ap[1];                      // ds_read_b64
      v2f b;
      const float* bp = w1c + (size_t)(kk + 2 * half) * FF;
      b[0] = bp[0]; b[1] = bp[FF];
      hacc = __builtin_amdgcn_wmma_f32_16x16x4_f32(false, a, false, b,
                                                   (short)0, hacc, false, false);
    }
    // bias + exact-erf GELU -> LDS chunk
    {
      const float bias = b1e[c0 + l16];
#pragma unroll
      for (int i = 0; i < 8; ++i) {
        const int mrow = i + 8 * half;
        const float v = hacc[i] + bias;
        const float g = 0.5f * v * (1.0f + erff(v * 0.70710678118654752f));
        hs[mrow * HS_STRIDE + wave * 16 + l16] = g;
      }
    }
    __syncthreads();

    // ---- stage 2: out(16 x 128, this wave) += h(16 x 128) @ w2 chunk ----
    const int oc0 = wave * 128;
    const float* w2c = w2e + (size_t)(ch * CHUNK) * D_MODEL + oc0 + l16;
    for (int kk = 0; kk < CHUNK; kk += 4) {
      v2f a;
      const float* ap = &hs[l16 * HS_STRIDE + kk + 2 * half];
      a[0] = ap[0]; a[1] = ap[1];
      const float* bp = w2c + (size_t)(kk + 2 * half) * D_MODEL;
#pragma unroll
      for (int j = 0; j < 8; ++j) {
        v2f b;
        b[0] = bp[j * 16];
        b[1] = bp[j * 16 + D_MODEL];
        oacc[j] = __builtin_amdgcn_wmma_f32_16x16x4_f32(false, a, false, b,
                                                        (short)0, oacc[j], false, false);
      }
    }
    __syncthreads();   // hs reused next chunk
  }

  // ---- epilogue: scale by routing weight, add b2, accumulate into out ----
  const int colbase = wave * 128;
#pragma unroll
  for (int j = 0; j < 8; ++j) {
    const int col = colbase + j * 16 + l16;
    const float bias = b2e[col];
#pragma unroll
    for (int i = 0; i < 8; ++i) {
      const int m = i + 8 * half;
      const float val = s_wt[m] * (oacc[j][i] + bias);
      unsafeAtomicAdd(&out[(size_t)s_tok[m] * D_MODEL + col], val);
    }
  }
}

extern "C" void kernel_launch(void* const* d_in, const int* in_sizes, int n_in,
                              void* d_out, int out_size, void* d_ws, size_t ws_size,
                              hipStream_t stream) {
  const float* x  = (const float*)d_in[0];
  const float* gw = (const float*)d_in[1];
  const float* w1 = (const float*)d_in[2];
  const float* b1 = (const float*)d_in[3];
  const float* w2 = (const float*)d_in[4];
  const float* b2 = (const float*)d_in[5];
  float* out = (float*)d_out;

  int*   counts = (int*)d_ws;
  int*   tok_id = (int*)((char*)d_ws + 256);
  float* tok_w  = (float*)((char*)d_ws + 256 + (size_t)NEXP * NTOK * sizeof(int));

  hipMemsetAsync(d_out, 0, (size_t)out_size * sizeof(float), stream);
  hipMemsetAsync(counts, 0, NEXP * sizeof(int), stream);

  moe_gate_kernel<<<NTOK / 8, 256, 0, stream>>>(x, gw, counts, tok_id, tok_w);

  dim3 grid(NTOK / TILE_M, NEXP);
  moe_expert_kernel<<<grid, 256, 0, stream>>>(x, w1, b1, w2, b2,
                                              counts, tok_id, tok_w, out);
}